// FastWeightLinearAttention_84387517432346
// MI455X (gfx1250) — compile-verified
//
#include <hip/hip_runtime.h>
#include <math.h>

// ---------------------------------------------------------------------------
// FastWeightLinearAttention (HOPE) for MI455X / gfx1250.
//  * fp32 matrix work via V_WMMA_F32_16X16X4_F32 (matches fp32 reference)
//  * serial fast-weight scan re-expressed as chunked causal linear attention
//    (C=64) so all heavy work is WMMA GEMMs
//  * global->LDS staging uses the CDNA5 async copy path
//    (global_load_async_to_lds_b128 + s_wait_asynccnt), double-buffered in
//    the two large 8192x1024x1024 GEMMs.
// ---------------------------------------------------------------------------

typedef __attribute__((ext_vector_type(2))) float v2f;
typedef __attribute__((ext_vector_type(8))) float v8f;

#define B_      4
#define T_      2048
#define DM_     1024
#define H_      16
#define HD_     64
#define CHUNK   64
#define NCHUNK  (T_ / CHUNK)          // 32
#define BH_     (B_ * H_)             // 64
#define MROWS   (B_ * T_)             // 8192

#define ASTR    20                    // LDS stride for 16-wide K panels of A
#define BSTR    72                    // LDS stride for 64-wide N panels of B
#define CSTR    76                    // LDS stride for 64x64 chunk tiles

static __device__ __forceinline__ v8f wmma4(v2f a, v2f b, v8f c) {
  // D = A(16x4,f32) * B(4x16,f32) + C(16x16,f32)
  return __builtin_amdgcn_wmma_f32_16x16x4_f32(false, a, false, b, (short)0, c,
                                               false, false);
}

static __device__ __forceinline__ v8f v8zero() {
  v8f z = {0.f, 0.f, 0.f, 0.f, 0.f, 0.f, 0.f, 0.f};
  return z;
}

// Async 16B global -> LDS copy (per-lane addresses, ASYNCcnt-tracked).
static __device__ __forceinline__ void async_ld16(void* lds, const void* g) {
  asm volatile("global_load_async_to_lds_b128 %0, %1, off"
               :
               : "v"((unsigned)(unsigned long long)lds),
                 "v"((unsigned long long)g)
               : "memory");
}
static __device__ __forceinline__ void wait_async(int keep) {
  if (keep == 0)
    asm volatile("s_wait_asynccnt 0x0" ::: "memory");
  else
    asm volatile("s_wait_asynccnt 0x3" ::: "memory");
}

// Fragment loaders (row-major, element stride).  lm = lane&15, hi = lane>>4.
// A[m][k]: m = m0+lm, k = k0 + 2*hi + c     B[k][n]: n = n0+lm, same k
static __device__ __forceinline__ v2f ldA(const float* s, int stride, int m0,
                                          int k0, int lm, int hi) {
  const float* p = s + (m0 + lm) * stride + k0 + 2 * hi;
  v2f a; a.x = p[0]; a.y = p[1]; return a;
}
static __device__ __forceinline__ v2f ldAT(const float* s, int stride, int m0,
                                           int k0, int lm, int hi) {
  int k = k0 + 2 * hi;                 // A[m][k] = src[k][m]
  v2f a;
  a.x = s[(k + 0) * stride + m0 + lm];
  a.y = s[(k + 1) * stride + m0 + lm];
  return a;
}
static __device__ __forceinline__ v2f ldB(const float* s, int stride, int n0,
                                          int k0, int lm, int hi) {
  int k = k0 + 2 * hi;
  v2f b;
  b.x = s[(k + 0) * stride + n0 + lm];
  b.y = s[(k + 1) * stride + n0 + lm];
  return b;
}
static __device__ __forceinline__ v2f ldBT(const float* s, int stride, int n0,
                                           int k0, int lm, int hi) {
  const float* p = s + (n0 + lm) * stride + k0 + 2 * hi;  // B[k][n] = src[n][k]
  v2f b; b.x = p[0]; b.y = p[1]; return b;
}

// ---------------------------------------------------------------------------
// Kernel 0: HOPE gains.  slow_state[b] -> LN -> W1 -> exact GELU -> W2 ->
// tanh+1.  gains[comp*64 + b*16 + h], comp 0=q,1=k,2=v.
// ---------------------------------------------------------------------------
__global__ __launch_bounds__(256) void gains_kernel(
    const float* __restrict__ sst, const float* __restrict__ ln_g,
    const float* __restrict__ ln_b, const float* __restrict__ w1,
    const float* __restrict__ b1, const float* __restrict__ w2,
    const float* __restrict__ b2, float* __restrict__ gains) {
  const int b = blockIdx.x;
  const int tid = threadIdx.x;
  __shared__ float red[256];
  __shared__ float xn[DM_];
  __shared__ float hb[256];
  const float* xrow = sst + (size_t)b * DM_;

  float part = 0.f;
  for (int i = tid; i < DM_; i += 256) part += xrow[i];
  red[tid] = part;
  __syncthreads();
  for (int s = 128; s > 0; s >>= 1) {
    if (tid < s) red[tid] += red[tid + s];
    __syncthreads();
  }
  const float mean = red[0] * (1.f / DM_);
  __syncthreads();
  part = 0.f;
  for (int i = tid; i < DM_; i += 256) {
    float d = xrow[i] - mean;
    part += d * d;
  }
  red[tid] = part;
  __syncthreads();
  for (int s = 128; s > 0; s >>= 1) {
    if (tid < s) red[tid] += red[tid + s];
    __syncthreads();
  }
  const float rstd = rsqrtf(red[0] * (1.f / DM_) + 1e-5f);
  __syncthreads();
  for (int i = tid; i < DM_; i += 256)
    xn[i] = (xrow[i] - mean) * rstd * ln_g[i] + ln_b[i];
  __syncthreads();

  float hsum = b1[tid];
  for (int i = 0; i < DM_; ++i) hsum += xn[i] * w1[i * 256 + tid];
  hb[tid] = 0.5f * hsum * (1.f + erff(hsum * 0.7071067811865475f));
  __syncthreads();

  if (tid < 3 * H_) {
    float p = b2[tid];
    for (int j = 0; j < 256; ++j) p += hb[j] * w2[j * (3 * H_) + tid];
    const int hh = tid / 3, comp = tid % 3;
    gains[comp * BH_ + b * H_ + hh] = tanhf(p) + 1.f;
  }
}

// ---------------------------------------------------------------------------
// Kernel 1: fused q/k/v projection (8192x1024x1024) + gain + (q,k) row norm.
// Block tile 128x64, 8 waves * (2x2 WMMA tiles).  Double-buffered async
// global->LDS staging: stage ib+1 streams while stage ib feeds the WMMAs.
// Output: dst[((b*16+h)*2048 + t)*64 + d]
// ---------------------------------------------------------------------------
__global__ __launch_bounds__(256) void qkv_gemm_kernel(
    const float* __restrict__ x, const float* __restrict__ wq,
    const float* __restrict__ wk, const float* __restrict__ wv,
    const float* __restrict__ gains, float* __restrict__ q_ws,
    float* __restrict__ k_ws, float* __restrict__ v_ws) {
  const int comp = blockIdx.z;
  const float* W = (comp == 0) ? wq : (comp == 1) ? wk : wv;
  float* dst = (comp == 0) ? q_ws : (comp == 1) ? k_ws : v_ws;

  const int m0g = blockIdx.x * 128;
  const int h = blockIdx.y;
  const int n0g = h * 64;
  const int b = m0g >> 11;
  const int tid = threadIdx.x;
  const int lane = tid & 31, lm = lane & 15, hi = lane >> 4, w = tid >> 5;

  __shared__ __align__(16) float As[2][128 * ASTR];  // 20.0 KB
  __shared__ __align__(16) float Bs[2][16 * BSTR];   //  9.0 KB
  __shared__ float Cs[128 * 65];                     // 33.3 KB
  __shared__ float scl[128];

  const int mt = (w & 3) * 2;
  const int nt = (w >> 2) * 2;

  // per-thread staging coordinates (3 async B128 copies per stage)
  const int ar0 = (tid) >> 2,        ac0 = ((tid) & 3) * 4;
  const int ar1 = (tid + 256) >> 2,  ac1 = ((tid + 256) & 3) * 4;
  const int br  = tid >> 4,          bc  = (tid & 15) * 4;

  v8f acc[2][2];
#pragma unroll
  for (int i = 0; i < 2; ++i)
#pragma unroll
    for (int j = 0; j < 2; ++j) acc[i][j] = v8zero();

  // prologue: stage 0
  async_ld16(&As[0][ar0 * ASTR + ac0], &x[(size_t)(m0g + ar0) * DM_ + ac0]);
  async_ld16(&As[0][ar1 * ASTR + ac1], &x[(size_t)(m0g + ar1) * DM_ + ac1]);
  async_ld16(&Bs[0][br * BSTR + bc], &W[(size_t)br * DM_ + n0g + bc]);

  const int NKB = DM_ / 16;  // 64
  for (int ib = 0; ib < NKB; ++ib) {
    const int buf = ib & 1;
    if (ib + 1 < NKB) {
      const int kb = (ib + 1) * 16;
      async_ld16(&As[buf ^ 1][ar0 * ASTR + ac0],
                 &x[(size_t)(m0g + ar0) * DM_ + kb + ac0]);
      async_ld16(&As[buf ^ 1][ar1 * ASTR + ac1],
                 &x[(size_t)(m0g + ar1) * DM_ + kb + ac1]);
      async_ld16(&Bs[buf ^ 1][br * BSTR + bc],
                 &W[(size_t)(kb + br) * DM_ + n0g + bc]);
      wait_async(3);                   // stage ib complete, ib+1 in flight
    } else {
      wait_async(0);
    }
    __syncthreads();                   // stage ib visible to all waves
    const float* Ab = As[buf];
    const float* Bb = Bs[buf];
#pragma unroll
    for (int kk = 0; kk < 16; kk += 4) {
      v2f a0 = ldA(Ab, ASTR, (mt + 0) * 16, kk, lm, hi);
      v2f a1 = ldA(Ab, ASTR, (mt + 1) * 16, kk, lm, hi);
      v2f b0 = ldB(Bb, BSTR, (nt + 0) * 16, kk, lm, hi);
      v2f b1 = ldB(Bb, BSTR, (nt + 1) * 16, kk, lm, hi);
      acc[0][0] = wmma4(a0, b0, acc[0][0]);
      acc[0][1] = wmma4(a0, b1, acc[0][1]);
      acc[1][0] = wmma4(a1, b0, acc[1][0]);
      acc[1][1] = wmma4(a1, b1, acc[1][1]);
    }
    __syncthreads();                   // all reads of buf done before refill
  }

  // norm/gain epilogue via LDS tile
#pragma unroll
  for (int i = 0; i < 2; ++i)
#pragma unroll
    for (int j = 0; j < 2; ++j)
#pragma unroll
      for (int r = 0; r < 8; ++r) {
        int row = (mt + i) * 16 + r + 8 * hi;
        int col = (nt + j) * 16 + lm;
        Cs[row * 65 + col] = acc[i][j][r];
      }
  __syncthreads();

  const float g = gains[comp * BH_ + b * H_ + h];
  if (tid < 128) {
    if (comp < 2) {
      float ss = 0.f;
      for (int d = 0; d < 64; ++d) {
        float v = Cs[tid * 65 + d];
        ss += v * v;
      }
      scl[tid] = g / (g * sqrtf(ss) + 1e-6f);   // (q*g)/(||q*g||+1e-6)
    } else {
      scl[tid] = g;
    }
  }
  __syncthreads();

  const size_t base = ((size_t)(b * H_ + h)) * T_ * HD_;
  const int t0 = m0g & (T_ - 1);
#pragma unroll 4
  for (int i = 0; i < 32; ++i) {
    int idx = tid + i * 256;
    int row = idx >> 6, d = idx & 63;
    dst[base + (size_t)(t0 + row) * HD_ + d] = Cs[row * 65 + d] * scl[row];
  }
}

// ---------------------------------------------------------------------------
// Kernel 2 (P1): per-chunk state increment  A_c[e][d] = sum_t K[t][e]*V[t][d]
// 64x64x64 WMMA GEMM per (chunk, bh) block; async-staged LDS tiles.
// ---------------------------------------------------------------------------
__global__ __launch_bounds__(256) void chunk_kv_kernel(
    const float* __restrict__ k_ws, const float* __restrict__ v_ws,
    float* __restrict__ A_ws) {
  const int c = blockIdx.x, bh = blockIdx.y;
  const int tid = threadIdx.x;
  const int lane = tid & 31, lm = lane & 15, hi = lane >> 4, w = tid >> 5;
  __shared__ __align__(16) float Ks[64 * CSTR];
  __shared__ __align__(16) float Vs[64 * CSTR];

  const size_t src = ((size_t)bh * T_ + c * CHUNK) * HD_;
#pragma unroll
  for (int i = 0; i < 4; ++i) {
    int lin = (tid + i * 256) * 4;
    int row = lin >> 6, col = lin & 63;
    async_ld16(&Ks[row * CSTR + col], &k_ws[src + lin]);
    async_ld16(&Vs[row * CSTR + col], &v_ws[src + lin]);
  }
  wait_async(0);
  __syncthreads();

  const int mt = w & 3;                 // e-tile
  const int ntb = (w >> 2) * 2;         // d-tile pair
  v8f acc[2];
  acc[0] = v8zero(); acc[1] = v8zero();
#pragma unroll
  for (int kk = 0; kk < 64; kk += 4) {
    v2f a = ldAT(Ks, CSTR, mt * 16, kk, lm, hi);   // K^T
    v2f b0 = ldB(Vs, CSTR, (ntb + 0) * 16, kk, lm, hi);
    v2f b1 = ldB(Vs, CSTR, (ntb + 1) * 16, kk, lm, hi);
    acc[0] = wmma4(a, b0, acc[0]);
    acc[1] = wmma4(a, b1, acc[1]);
  }

  const size_t dstb = ((size_t)bh * NCHUNK + c) * (HD_ * HD_);
#pragma unroll
  for (int j = 0; j < 2; ++j)
#pragma unroll
    for (int r = 0; r < 8; ++r) {
      int e = mt * 16 + r + 8 * hi;
      int d = (ntb + j) * 16 + lm;
      A_ws[dstb + e * HD_ + d] = acc[j][r];
    }
}

// ---------------------------------------------------------------------------
// Kernel 3 (P2): in-place exclusive prefix over 32 chunk increments per
// (b,h); seeds with memory^T, emits mem_final (transposed) to d_out.
// ---------------------------------------------------------------------------
__global__ __launch_bounds__(256) void prefix_kernel(
    const float* __restrict__ mem0, float* __restrict__ A_ws,
    float* __restrict__ mem_out) {
  const int bh = blockIdx.x;
  const int tid = threadIdx.x;
  float accv[16];
#pragma unroll
  for (int i = 0; i < 16; ++i) {
    int lin = tid + i * 256;
    int e = lin >> 6, d = lin & 63;
    accv[i] = mem0[(size_t)bh * 4096 + d * 64 + e];  // S0 = mem0^T
  }
  for (int c = 0; c < NCHUNK; ++c) {
    const size_t base = ((size_t)bh * NCHUNK + c) * 4096;
#pragma unroll
    for (int i = 0; i < 16; ++i) {
      int lin = tid + i * 256;
      float t = A_ws[base + lin];
      A_ws[base + lin] = accv[i];        // state before chunk c
      accv[i] += t;
    }
  }
#pragma unroll
  for (int i = 0; i < 16; ++i) {
    int lin = tid + i * 256;
    int e = lin >> 6, d = lin & 63;
    mem_out[(size_t)bh * 4096 + d * 64 + e] = accv[i];
  }
}

// ---------------------------------------------------------------------------
// Kernel 4 (P3): per-chunk output  S = QK^T (causal), Y = S V + Q M_prefix.
// Three 64^3 WMMA GEMMs per block; masked S round-trips through the K buffer.
// Y written directly in [b][t][dm] layout for the final projection.
// ---------------------------------------------------------------------------
__global__ __launch_bounds__(256) void chunk_attn_kernel(
    const float* __restrict__ q_ws, const float* __restrict__ k_ws,
    const float* __restrict__ v_ws, const float* __restrict__ M_ws,
    float* __restrict__ Y_ws) {
  const int c = blockIdx.x, bh = blockIdx.y;
  const int b = bh >> 4, h = bh & 15;
  const int tid = threadIdx.x;
  const int lane = tid & 31, lm = lane & 15, hi = lane >> 4, w = tid >> 5;
  __shared__ __align__(16) float Qs[64 * CSTR];
  __shared__ __align__(16) float KSs[64 * CSTR];   // K, then masked S
  __shared__ __align__(16) float Vs[64 * CSTR];

  const size_t src = ((size_t)bh * T_ + c * CHUNK) * HD_;
#pragma unroll
  for (int i = 0; i < 4; ++i) {
    int lin = (tid + i * 256) * 4;
    int row = lin >> 6, col = lin & 63;
    async_ld16(&Qs[row * CSTR + col], &q_ws[src + lin]);
    async_ld16(&KSs[row * CSTR + col], &k_ws[src + lin]);
    async_ld16(&Vs[row * CSTR + col], &v_ws[src + lin]);
  }
  wait_async(0);
  __syncthreads();

  const int mt = w & 3;                 // t1-tile
  const int ntb = (w >> 2) * 2;         // t2/d tile pair
  v8f s[2];
  s[0] = v8zero(); s[1] = v8zero();
#pragma unroll
  for (int kk = 0; kk < 64; kk += 4) {  // contract over e
    v2f a = ldA(Qs, CSTR, mt * 16, kk, lm, hi);
    v2f b0 = ldBT(KSs, CSTR, (ntb + 0) * 16, kk, lm, hi);  // K^T
    v2f b1 = ldBT(KSs, CSTR, (ntb + 1) * 16, kk, lm, hi);
    s[0] = wmma4(a, b0, s[0]);
    s[1] = wmma4(a, b1, s[1]);
  }
#pragma unroll
  for (int j = 0; j < 2; ++j)           // causal mask incl. diagonal
#pragma unroll
    for (int r = 0; r < 8; ++r) {
      int t1 = mt * 16 + r + 8 * hi;
      int t2 = (ntb + j) * 16 + lm;
      if (t2 > t1) s[j][r] = 0.f;
    }
  __syncthreads();                      // all waves done reading K
#pragma unroll
  for (int j = 0; j < 2; ++j)
#pragma unroll
    for (int r = 0; r < 8; ++r) {
      int t1 = mt * 16 + r + 8 * hi;
      int t2 = (ntb + j) * 16 + lm;
      KSs[t1 * CSTR + t2] = s[j][r];
    }
  __syncthreads();

  const float* Mp = M_ws + ((size_t)bh * NCHUNK + c) * 4096;  // [e][d]
  v8f acc[2];
  acc[0] = v8zero(); acc[1] = v8zero();
#pragma unroll
  for (int kk = 0; kk < 64; kk += 4) {  // inter-chunk: Q @ M_prefix
    v2f a = ldA(Qs, CSTR, mt * 16, kk, lm, hi);
    int k = kk + 2 * hi;
    v2f b0, b1;
    b0.x = Mp[(k + 0) * 64 + (ntb + 0) * 16 + lm];
    b0.y = Mp[(k + 1) * 64 + (ntb + 0) * 16 + lm];
    b1.x = Mp[(k + 0) * 64 + (ntb + 1) * 16 + lm];
    b1.y = Mp[(k + 1) * 64 + (ntb + 1) * 16 + lm];
    acc[0] = wmma4(a, b0, acc[0]);
    acc[1] = wmma4(a, b1, acc[1]);
  }
#pragma unroll
  for (int kk = 0; kk < 64; kk += 4) {  // intra-chunk: tril(S) @ V
    v2f a = ldA(KSs, CSTR, mt * 16, kk, lm, hi);
    v2f b0 = ldB(Vs, CSTR, (ntb + 0) * 16, kk, lm, hi);
    v2f b1 = ldB(Vs, CSTR, (ntb + 1) * 16, kk, lm, hi);
    acc[0] = wmma4(a, b0, acc[0]);
    acc[1] = wmma4(a, b1, acc[1]);
  }

  const size_t rowb = (size_t)b * T_ + c * CHUNK;
#pragma unroll
  for (int j = 0; j < 2; ++j)
#pragma unroll
    for (int r = 0; r < 8; ++r) {
      int t = mt * 16 + r + 8 * hi;
      int d = (ntb + j) * 16 + lm;
      Y_ws[(rowb + t) * DM_ + h * 64 + d] = acc[j][r];
    }
}

// ---------------------------------------------------------------------------
// Kernel 5 (P4): out = Y @ out_w + out_b  (8192x1024x1024), same async
// double-buffered WMMA tiling, bias epilogue straight to d_out.
// ---------------------------------------------------------------------------
__global__ __launch_bounds__(256) void out_gemm_kernel(
    const float* __restrict__ Y, const float* __restrict__ Wo,
    const float* __restrict__ bias, float* __restrict__ out) {
  const int m0g = blockIdx.x * 128;
  const int n0g = blockIdx.y * 64;
  const int tid = threadIdx.x;
  const int lane = tid & 31, lm = lane & 15, hi = lane >> 4, w = tid >> 5;

  __shared__ __align__(16) float As[2][128 * ASTR];
  __shared__ __align__(16) float Bs[2][16 * BSTR];

  const int mt = (w & 3) * 2;
  const int nt = (w >> 2) * 2;

  const int ar0 = (tid) >> 2,        ac0 = ((tid) & 3) * 4;
  const int ar1 = (tid + 256) >> 2,  ac1 = ((tid + 256) & 3) * 4;
  const int br  = tid >> 4,          bc  = (tid & 15) * 4;

  v8f acc[2][2];
#pragma unroll
  for (int i = 0; i < 2; ++i)
#pragma unroll
    for (int j = 0; j < 2; ++j) acc[i][j] = v8zero();

  async_ld16(&As[0][ar0 * ASTR + ac0], &Y[(size_t)(m0g + ar0) * DM_ + ac0]);
  async_ld16(&As[0][ar1 * ASTR + ac1], &Y[(size_t)(m0g + ar1) * DM_ + ac1]);
  async_ld16(&Bs[0][br * BSTR + bc], &Wo[(size_t)br * DM_ + n0g + bc]);

  const int NKB = DM_ / 16;
  for (int ib = 0; ib < NKB; ++ib) {
    const int buf = ib & 1;
    if (ib + 1 < NKB) {
      const int kb = (ib + 1) * 16;
      async_ld16(&As[buf ^ 1][ar0 * ASTR + ac0],
                 &Y[(size_t)(m0g + ar0) * DM_ + kb + ac0]);
      async_ld16(&As[buf ^ 1][ar1 * ASTR + ac1],
                 &Y[(size_t)(m0g + ar1) * DM_ + kb + ac1]);
      async_ld16(&Bs[buf ^ 1][br * BSTR + bc],
                 &Wo[(size_t)(kb + br) * DM_ + n0g + bc]);
      wait_async(3);
    } else {
      wait_async(0);
    }
    __syncthreads();
    const float* Ab = As[buf];
    const float* Bb = Bs[buf];
#pragma unroll
    for (int kk = 0; kk < 16; kk += 4) {
      v2f a0 = ldA(Ab, ASTR, (mt + 0) * 16, kk, lm, hi);
      v2f a1 = ldA(Ab, ASTR, (mt + 1) * 16, kk, lm, hi);
      v2f b0 = ldB(Bb, BSTR, (nt + 0) * 16, kk, lm, hi);
      v2f b1 = ldB(Bb, BSTR, (nt + 1) * 16, kk, lm, hi);
      acc[0][0] = wmma4(a0, b0, acc[0][0]);
      acc[0][1] = wmma4(a0, b1, acc[0][1]);
      acc[1][0] = wmma4(a1, b0, acc[1][0]);
      acc[1][1] = wmma4(a1, b1, acc[1][1]);
    }
    __syncthreads();
  }

#pragma unroll
  for (int i = 0; i < 2; ++i)
#pragma unroll
    for (int j = 0; j < 2; ++j)
#pragma unroll
      for (int r = 0; r < 8; ++r) {
        int row = m0g + (mt + i) * 16 + r + 8 * hi;
        int col = n0g + (nt + j) * 16 + lm;
        out[(size_t)row * DM_ + col] = acc[i][j][r] + bias[col];
      }
}

// ---------------------------------------------------------------------------
extern "C" void kernel_launch(void* const* d_in, const int* in_sizes, int n_in,
                              void* d_out, int out_size, void* d_ws,
                              size_t ws_size, hipStream_t stream) {
  (void)in_sizes; (void)n_in; (void)out_size; (void)ws_size;
  const float* x    = (const float*)d_in[0];
  const float* sst  = (const float*)d_in[1];
  const float* mem0 = (const float*)d_in[2];
  const float* wq   = (const float*)d_in[3];
  const float* wk   = (const float*)d_in[4];
  const float* wv   = (const float*)d_in[5];
  const float* lng  = (const float*)d_in[6];
  const float* lnb  = (const float*)d_in[7];
  const float* hw1  = (const float*)d_in[8];
  const float* hb1  = (const float*)d_in[9];
  const float* hw2  = (const float*)d_in[10];
  const float* hb2  = (const float*)d_in[11];
  const float* outw = (const float*)d_in[12];
  const float* outb = (const float*)d_in[13];

  float* out = (float*)d_out;                           // [4][2048][1024]
  float* mem_out = out + (size_t)B_ * T_ * DM_;         // [4][16][64][64]

  const size_t PROJ = (size_t)BH_ * T_ * HD_;           // 8,388,608 floats
  float* ws    = (float*)d_ws;
  float* gains = ws;                                    // 256 (192 used)
  float* q_ws  = ws + 256;
  float* k_ws  = q_ws + PROJ;
  float* v_ws  = k_ws + PROJ;
  float* A_ws  = v_ws + PROJ;                           // chunk sums -> prefix
  float* Y_ws  = A_ws + (size_t)BH_ * NCHUNK * HD_ * HD_;

  gains_kernel<<<dim3(B_), dim3(256), 0, stream>>>(sst, lng, lnb, hw1, hb1,
                                                   hw2, hb2, gains);
  qkv_gemm_kernel<<<dim3(MROWS / 128, H_, 3), dim3(256), 0, stream>>>(
      x, wq, wk, wv, gains, q_ws, k_ws, v_ws);
  chunk_kv_kernel<<<dim3(NCHUNK, BH_), dim3(256), 0, stream>>>(k_ws, v_ws,
                                                               A_ws);
  prefix_kernel<<<dim3(BH_), dim3(256), 0, stream>>>(mem0, A_ws, mem_out);
  chunk_attn_kernel<<<dim3(NCHUNK, BH_), dim3(256), 0, stream>>>(
      q_ws, k_ws, v_ws, A_ws, Y_ws);
  out_gemm_kernel<<<dim3(MROWS / 128, DM_ / 64), dim3(256), 0, stream>>>(
      Y_ws, outw, outb, out);
}